// GraphNorm_24275155157125
// MI455X (gfx1250) — compile-verified
//
#include <hip/hip_runtime.h>

#define NCH   128
#define NB    512
#define EPSV  1e-5f
#define CHUNK 512

typedef __attribute__((ext_vector_type(2))) float v2f;
typedef __attribute__((ext_vector_type(4))) float v4f;
typedef __attribute__((ext_vector_type(8))) float v8f;

// ---------------------------------------------------------------------------
// Per-segment node counts. batch is sorted, so run-length encode locally and
// flush one atomic per run (~2 atomics / 64 rows).
// ---------------------------------------------------------------------------
__global__ void gn_count(const int* __restrict__ batch, float* __restrict__ gcnt, int n) {
    long t = (long)blockIdx.x * blockDim.x + threadIdx.x;
    long base = t * 64;
    if (base >= n) return;
    int cur = -1;
    float run = 0.0f;
    for (int j = 0; j < 64; ++j) {
        long r = base + j;
        if (r >= n) break;
        int s = batch[r];
        if (s != cur) {
            if (run > 0.0f && cur >= 0 && cur < NB) atomicAdd(&gcnt[cur], run);
            cur = s;
            run = 0.0f;
        }
        run += 1.0f;
    }
    if (run > 0.0f && cur >= 0 && cur < NB) atomicAdd(&gcnt[cur], run);
}

// ---------------------------------------------------------------------------
// One-pass segment sum + sum-of-squares via V_WMMA_F32_16X16X4_F32.
//   D(16 slots x 16 ch) += A(16x4 one-hot slot selector) * B(4 rows x 16 ch)
// Block = 256 threads = 8 waves; wave w owns channels [16w, 16w+16).
// Each block consumes CHUNK=512 contiguous sorted rows.
// Full chunks (all but the last block) take a guard-free streaming path.
// ---------------------------------------------------------------------------
__global__ __launch_bounds__(256) void gn_stats(const float* __restrict__ x,
                                                const int* __restrict__ batch,
                                                float* __restrict__ gsum,
                                                float* __restrict__ gsumsq,
                                                int n) {
    __shared__ int s_slot[CHUNK];
    const int tid = threadIdx.x;
    const int chunkStart = blockIdx.x * CHUNK;
    const int segBase = batch[chunkStart];
    const bool fullChunk = (chunkStart + CHUNK) <= n;   // scalar (uniform) condition

    for (int i = tid; i < CHUNK; i += 256) {
        int r = chunkStart + i;
        int s = (r < n) ? (batch[r] - segBase) : -1;    // -1 => A column all zero
        s_slot[i] = (s > 15) ? 15 : s;                  // 16 slots ample for 512 sorted rows
    }
    __syncthreads();

    const int wave   = tid >> 5;
    const int lane   = tid & 31;
    const int half   = lane >> 4;   // half-wave selects K pair / slot group
    const int lm     = lane & 15;
    const int chBase = wave * 16;
    const int kOff   = half * 2;

    v8f accS = {};  // segment sum(x)
    v8f accQ = {};  // segment sum(x^2)

    if (fullChunk) {
        // Guard-free streaming: 2 plain dword loads + 2 WMMAs per 4 rows.
        const float* __restrict__ p =
            x + (size_t)(chunkStart + kOff) * NCH + chBase + lm;
#pragma unroll 4
        for (int r = 0; r < CHUNK; r += 4) {
            const int sA = s_slot[r + kOff];
            const int sB = s_slot[r + kOff + 1];
            v2f a, b, q;
            a.x = (sA == lm) ? 1.0f : 0.0f;
            a.y = (sB == lm) ? 1.0f : 0.0f;
            const float bx = __builtin_nontemporal_load(p);
            const float by = __builtin_nontemporal_load(p + NCH);
            b.x = bx;      b.y = by;
            q.x = bx * bx; q.y = by * by;
            accS = __builtin_amdgcn_wmma_f32_16x16x4_f32(false, a, false, b, (short)0,
                                                         accS, false, false);
            accQ = __builtin_amdgcn_wmma_f32_16x16x4_f32(false, a, false, q, (short)0,
                                                         accQ, false, false);
            p += 4 * NCH;
        }
    } else {
        // Tail chunk only: per-row guards.
        for (int r = 0; r < CHUNK; r += 4) {
            const int sA = s_slot[r + kOff];
            const int sB = s_slot[r + kOff + 1];
            v2f a, b, q;
            a.x = (sA == lm) ? 1.0f : 0.0f;
            a.y = (sB == lm) ? 1.0f : 0.0f;
            const long rA = (long)chunkStart + r + kOff;
            const float bx = (rA     < n) ? x[rA * NCH + chBase + lm]       : 0.0f;
            const float by = (rA + 1 < n) ? x[(rA + 1) * NCH + chBase + lm] : 0.0f;
            b.x = bx;      b.y = by;
            q.x = bx * bx; q.y = by * by;
            accS = __builtin_amdgcn_wmma_f32_16x16x4_f32(false, a, false, b, (short)0,
                                                         accS, false, false);
            accQ = __builtin_amdgcn_wmma_f32_16x16x4_f32(false, a, false, q, (short)0,
                                                         accQ, false, false);
        }
    }

    // Flush only touched slots (untouched accumulators are exactly 0.0).
#pragma unroll
    for (int v = 0; v < 8; ++v) {
        const float sv = accS[v];
        const float qv = accQ[v];
        if (sv != 0.0f || qv != 0.0f) {
            int seg = segBase + v + 8 * half;
            if (seg > NB - 1) seg = NB - 1;
            const int off = seg * NCH + chBase + lm;
            atomicAdd(&gsum[off],   sv);
            atomicAdd(&gsumsq[off], qv);
        }
    }
}

// ---------------------------------------------------------------------------
// Analytic finalize:
//   mean = sum/cnt,  var = E[x^2] - mean^2 * ms * (2 - f*ms),  f = cnt>0 ? 1 : 0
//   scaledMean = mean*ms,   outScale = weight / sqrt(var + eps)
// ---------------------------------------------------------------------------
__global__ void gn_finalize(const float* __restrict__ gsum,
                            const float* __restrict__ gsumsq,
                            const float* __restrict__ gcnt,
                            const float* __restrict__ weight,
                            const float* __restrict__ mean_scale,
                            float* __restrict__ sm,
                            float* __restrict__ os) {
    const int i = blockIdx.x * blockDim.x + threadIdx.x;
    if (i >= NB * NCH) return;
    const int seg = i >> 7;
    const int c   = i & (NCH - 1);
    const float cnt = gcnt[seg];
    const float inv = 1.0f / fmaxf(cnt, 1.0f);
    const float f   = cnt * inv;  // 1 for non-empty, 0 for empty segment
    const float mean = gsum[i] * inv;
    const float e2   = gsumsq[i] * inv;
    const float ms   = mean_scale[c];
    float var = e2 - mean * mean * ms * (2.0f - f * ms);
    var = fmaxf(var, 0.0f);
    sm[i] = mean * ms;
    os[i] = weight[c] / sqrtf(var + EPSV);
}

// ---------------------------------------------------------------------------
// Streaming normalize: out = (x - sm[seg]) * os[seg] + bias.
// x and out are non-temporal (no reuse, 512 MB combined); batch + per-segment
// tables stay L2-resident.
// ---------------------------------------------------------------------------
__global__ __launch_bounds__(256) void gn_apply(const v4f* __restrict__ x4,
                                                const int* __restrict__ batch,
                                                const v4f* __restrict__ sm4,
                                                const v4f* __restrict__ os4,
                                                const v4f* __restrict__ bias4,
                                                v4f* __restrict__ out4,
                                                int n) {
    const long idx = (long)blockIdx.x * 256 + threadIdx.x;
    const long total = (long)n * (NCH / 4);
    if (idx >= total) return;
    const int row = (int)(idx >> 5);     // NCH/4 == 32 v4f per row
    const int cg  = (int)(idx & 31);
    int seg = batch[row];
    seg = seg < 0 ? 0 : (seg > NB - 1 ? NB - 1 : seg);
    const v4f xv = __builtin_nontemporal_load(&x4[idx]);
    const v4f m  = sm4[(long)seg * 32 + cg];
    const v4f s  = os4[(long)seg * 32 + cg];
    const v4f b  = bias4[cg];
    const v4f o  = (xv - m) * s + b;
    __builtin_nontemporal_store(o, &out4[idx]);
}

// ---------------------------------------------------------------------------
// Workspace layout (floats):
//   [0, 65536)            gsum
//   [65536, 131072)       gsumsq
//   [131072, 131584)      gcnt
//   [131584, 197120)      scaledMean
//   [197120, 262656)      outScale
// ---------------------------------------------------------------------------
extern "C" void kernel_launch(void* const* d_in, const int* in_sizes, int n_in,
                              void* d_out, int out_size, void* d_ws, size_t ws_size,
                              hipStream_t stream) {
    const float* x      = (const float*)d_in[0];
    const int*   batch  = (const int*)d_in[1];
    const float* weight = (const float*)d_in[2];
    const float* bias   = (const float*)d_in[3];
    const float* mscale = (const float*)d_in[4];
    const int n = in_sizes[0] / NCH;

    float* ws     = (float*)d_ws;
    float* gsum   = ws;
    float* gsumsq = ws + NB * NCH;
    float* gcnt   = ws + 2 * NB * NCH;
    float* sm     = ws + 2 * NB * NCH + NB;
    float* os     = sm + NB * NCH;

    hipMemsetAsync(ws, 0, (size_t)(2 * NB * NCH + NB) * sizeof(float), stream);

    const int cntThreads = (n + 63) / 64;
    gn_count<<<(cntThreads + 255) / 256, 256, 0, stream>>>(batch, gcnt, n);

    gn_stats<<<(n + CHUNK - 1) / CHUNK, 256, 0, stream>>>(x, batch, gsum, gsumsq, n);

    gn_finalize<<<(NB * NCH + 255) / 256, 256, 0, stream>>>(gsum, gsumsq, gcnt,
                                                            weight, mscale, sm, os);

    const long total4 = (long)n * (NCH / 4);
    gn_apply<<<(int)((total4 + 255) / 256), 256, 0, stream>>>(
        (const v4f*)x, batch, (const v4f*)sm, (const v4f*)os,
        (const v4f*)bias, (v4f*)d_out, n);
}